// SigNet_49349174231407
// MI455X (gfx1250) — compile-verified
//
#include <hip/hip_runtime.h>

// Problem constants (match reference)
#define BATCH  256
#define LPTS   256
#define CIN    8
#define CCH    9            // CIN + time channel
#define SIGCH  819          // 9 + 81 + 729
#define KPAD   832          // SIGCH padded to multiple of 4 (and 64B rows)
#define NOUT   128

typedef __attribute__((ext_vector_type(2))) float v2f;
typedef __attribute__((ext_vector_type(8))) float v8f;

// ---------------------------------------------------------------------------
// Kernel 1: depth-3 path signature via sequential Chen recursion.
// One workgroup (256 threads, 8 wave32) per batch element.
// s1 (9) and s2 (81) live in LDS; the 729 s3 accumulators live in registers,
// 3 per thread. Writes zero-padded signature rows [B][KPAD] to workspace.
// ---------------------------------------------------------------------------
__global__ __launch_bounds__(256) void signet_sig_kernel(
    const float* __restrict__ inp,   // [B, L, CIN]
    float* __restrict__ sig)         // [B, KPAD]
{
    const int b = blockIdx.x;
    const int t = threadIdx.x;

    __shared__ float sh_dx[CCH];
    __shared__ float sh_s1[CCH];
    __shared__ float sh_s2[CCH * CCH];

    // Static ownership of s3 elements: idx = t, t+256, t+512 (valid if < 729)
    int i0, j0, k0, i1, j1, k1;
    int i2 = -1, j2 = 0, k2 = 0;
    { const int idx = t;        i0 = idx / 81; j0 = (idx / 9) % 9; k0 = idx % 9; }
    { const int idx = t + 256;  i1 = idx / 81; j1 = (idx / 9) % 9; k1 = idx % 9; }
    { const int idx = t + 512;  if (idx < 729) { i2 = idx / 81; j2 = (idx / 9) % 9; k2 = idx % 9; } }
    float a0 = 0.f, a1 = 0.f, a2 = 0.f;

    // s2 ownership (threads 0..80)
    const int s2i = t / 9;
    const int s2j = t - 9 * s2i;

    if (t < CCH)        sh_s1[t] = 0.f;
    if (t < CCH * CCH)  sh_s2[t] = 0.f;
    float xprev = 0.f;                 // basepoint = 0 for all channels
    __syncthreads();

    const float* ip = inp + (size_t)b * LPTS * CIN;

    for (int l = 0; l < LPTS; ++l) {
        // --- phase 1: segment increment dx into LDS (threads 0..8) ---------
        if (t < CCH) {
            const float cur = (t == 0) ? (float)l * (1.0f / 255.0f)
                                       : ip[l * CIN + (t - 1)];
            sh_dx[t] = cur - xprev;
            xprev = cur;
        }
        __syncthreads();

        // --- phase 2: s3 += e3 + s2_old (x) dx + s1_old (x) e2 -------------
        {
            const float dxi = sh_dx[i0], dxj = sh_dx[j0], dxk = sh_dx[k0];
            const float s2o = sh_s2[i0 * 9 + j0], s1o = sh_s1[i0];
            a0 += dxk * (dxi * dxj * (1.0f / 6.0f) + s2o + 0.5f * s1o * dxj);
        }
        {
            const float dxi = sh_dx[i1], dxj = sh_dx[j1], dxk = sh_dx[k1];
            const float s2o = sh_s2[i1 * 9 + j1], s1o = sh_s1[i1];
            a1 += dxk * (dxi * dxj * (1.0f / 6.0f) + s2o + 0.5f * s1o * dxj);
        }
        if (i2 >= 0) {
            const float dxi = sh_dx[i2], dxj = sh_dx[j2], dxk = sh_dx[k2];
            const float s2o = sh_s2[i2 * 9 + j2], s1o = sh_s1[i2];
            a2 += dxk * (dxi * dxj * (1.0f / 6.0f) + s2o + 0.5f * s1o * dxj);
        }
        // cache old s1/s2 values needed for the s2/s1 updates
        float s2self = 0.f, s1row = 0.f, s1self = 0.f;
        if (t < 81) { s2self = sh_s2[t]; s1row = sh_s1[s2i]; }
        if (t < 9)  { s1self = sh_s1[t]; }
        __syncthreads();

        // --- phase 3: s2 += e2 + s1_old (x) dx ; s1 += dx ------------------
        if (t < 81) {
            const float dxi = sh_dx[s2i], dxj = sh_dx[s2j];
            sh_s2[t] = s2self + dxj * (0.5f * dxi + s1row);
        }
        if (t < 9) {
            sh_s1[t] = s1self + sh_dx[t];
        }
        __syncthreads();
    }

    // --- emit zero-padded signature row ------------------------------------
    float* sg = sig + (size_t)b * KPAD;
    if (t < 9)   sg[t] = sh_s1[t];
    if (t < 81)  sg[9 + t] = sh_s2[t];
    sg[90 + t] = a0;                   // idx 0..255
    sg[90 + t + 256] = a1;             // idx 256..511
    if (i2 >= 0) sg[90 + t + 512] = a2;
    if (t >= 81 && t < 81 + (KPAD - SIGCH)) sg[SIGCH + (t - 81)] = 0.f;
}

// ---------------------------------------------------------------------------
// Kernel 2: copy W [819,128] into zero-padded W_pad [832,128]
// ---------------------------------------------------------------------------
__global__ void signet_wpad_kernel(const float* __restrict__ W,
                                   float* __restrict__ Wp)
{
    const int idx = blockIdx.x * blockDim.x + threadIdx.x;
    if (idx >= KPAD * NOUT) return;
    const int row = idx / NOUT;
    Wp[idx] = (row < SIGCH) ? W[idx] : 0.f;
}

// ---------------------------------------------------------------------------
// Kernel 3: out[256,128] = sig_pad[256,832] @ W_pad[832,128] + bias
// using V_WMMA_F32_16X16X4_F32. One wave32 per 16x16 output tile.
// A frag (16x4 f32): lanes 0-15 hold K={k,k+1}, lanes 16-31 hold K={k+2,k+3},
// M = lane%16. B frag mirrors in K. C/D: VGPR r -> M = r + 8*(lane/16),
// N = lane%16.
// ---------------------------------------------------------------------------
__global__ __launch_bounds__(128) void signet_gemm_wmma(
    const float* __restrict__ A,     // [256, KPAD]
    const float* __restrict__ Bm,    // [KPAD, 128]
    const float* __restrict__ bias,  // [128]
    float* __restrict__ out)         // [256, 128]
{
    const int wave = (blockIdx.x * blockDim.x + threadIdx.x) >> 5;  // 0..127
    const int lane = threadIdx.x & 31;
    const int half = lane >> 4;     // 0: K pair {0,1}, 1: K pair {2,3}
    const int lo   = lane & 15;

    const int tm = wave >> 3;       // 0..15 (M tiles)
    const int tn = wave & 7;        // 0..7  (N tiles)

    const int arow = tm * 16 + lo;
    const int bcol = tn * 16 + lo;

    const float* Ar = A + (size_t)arow * KPAD;

    v8f c = {};
    for (int k = 0; k < KPAD; k += 4) {
        const int ka = k + 2 * half;
        v2f a, bfrag;
        a.x     = Ar[ka];
        a.y     = Ar[ka + 1];
        bfrag.x = Bm[(size_t)ka * NOUT + bcol];
        bfrag.y = Bm[(size_t)(ka + 1) * NOUT + bcol];
        // 8 args: (neg_a, A, neg_b, B, c_mod, C, reuse_a, reuse_b)
        c = __builtin_amdgcn_wmma_f32_16x16x4_f32(
            false, a, false, bfrag, (short)0, c, false, false);
    }

    const float bv = bias[bcol];
#pragma unroll
    for (int r = 0; r < 8; ++r) {
        const int orow = tm * 16 + 8 * half + r;
        out[(size_t)orow * NOUT + bcol] = c[r] + bv;
    }
}

// ---------------------------------------------------------------------------
extern "C" void kernel_launch(void* const* d_in, const int* in_sizes, int n_in,
                              void* d_out, int out_size, void* d_ws, size_t ws_size,
                              hipStream_t stream)
{
    const float* inp  = (const float*)d_in[0];  // [256,256,8]
    const float* W    = (const float*)d_in[1];  // [819,128]
    const float* bias = (const float*)d_in[2];  // [128]
    float* out        = (float*)d_out;          // [256,128]

    // Workspace layout
    float* sig_pad = (float*)d_ws;                                   // 256*832 f32
    float* w_pad   = (float*)((char*)d_ws + (size_t)BATCH * KPAD * sizeof(float));

    // 1) signatures (one workgroup per batch)
    signet_sig_kernel<<<BATCH, 256, 0, stream>>>(inp, sig_pad);

    // 2) zero-padded weight copy
    {
        const int total = KPAD * NOUT;
        signet_wpad_kernel<<<(total + 255) / 256, 256, 0, stream>>>(W, w_pad);
    }

    // 3) WMMA GEMM + bias: 128 tiles, 4 waves/block -> 32 blocks x 128 threads
    signet_gemm_wmma<<<32, 128, 0, stream>>>(sig_pad, w_pad, bias, out);
}